// RWN_16329465659691
// MI455X (gfx1250) — compile-verified
//
#include <hip/hip_runtime.h>

// RWN on MI455X (gfx1250, wave32).
// Pass 0: zero the A accumulator (workspace) — required every call (atomics).
// Pass 1: per-channel windowed softmax weights accumulated into A[B][N][N][49].
//   exp(|a-b|) = max(exp(a)exp(-b), exp(-a)exp(b)) -> exp(+/-v) cached in LDS
//   per 22x22 halo tile. K=131 split into 8 groups/tile for occupancy; partial
//   sums combined with global_atomic_add_f32 (mask folded into the add).
// Pass 2: y = W^T x2 as banded GEMM via V_WMMA_F32_16X16X4_F32,
//   one wave per 16-pixel row segment, Y[16x21] = sum_r M_r[16x22] * X_r[22x21].
//   Operand gathers are branchless (clamped address + cndmask zero-select).

#define RR   3
#define NN   128
#define KK   131
#define CC   21
#define WW   49
#define TILE 16
#define HALO 22   // TILE + 2*RR
#define NPIX (NN * NN)
#define NGRP 8    // channel groups per tile in pass 1
#define KG   17   // ceil(131 / 8)

typedef __attribute__((ext_vector_type(2))) float v2f;
typedef __attribute__((ext_vector_type(8))) float v8f;

__device__ __forceinline__ float fexp(float x) {
  // exp(x) = 2^(x * log2(e)); lowers to v_exp_f32
  return __builtin_amdgcn_exp2f(x * 1.44269504088896340736f);
}

// ---------------------------------------------------------------------------
// Pass 0: zero A accumulator. 2*128*128*49 floats = 1,605,632 = 6272 * 256.
// ---------------------------------------------------------------------------
__global__ __launch_bounds__(256)
void rwn_zero(float* __restrict__ A) {
  A[(size_t)blockIdx.x * 256 + threadIdx.x] = 0.0f;
}

// ---------------------------------------------------------------------------
// Pass 1: A[b][i][j][w] += m(i,j) * sum_{k in group} p_k * e_kw / Z_k
// ---------------------------------------------------------------------------
__global__ __launch_bounds__(256)
void rwn_pass1(const float* __restrict__ feats,   // [B][K][N][N]
               const int*   __restrict__ mask,    // [B][N][N]
               const float* __restrict__ convp,   // [K]
               float*       __restrict__ A)       // [B][N][N][W] (accumulated)
{
  __shared__ float2 tile[HALO * HALO];            // (exp(v), exp(-v)) per halo px

  const int b  = blockIdx.z / NGRP;
  const int g  = blockIdx.z % NGRP;
  const int i0 = blockIdx.y * TILE;
  const int j0 = blockIdx.x * TILE;
  const int t  = threadIdx.x;
  const int ty = t >> 4, tx = t & 15;
  const int i = i0 + ty, j = j0 + tx;

  const int k0 = g * KG;
  const int k1 = (k0 + KG < KK) ? (k0 + KG) : KK;

  bool rv[7], cv[7];
  #pragma unroll
  for (int d = 0; d < 7; ++d) {
    rv[d] = (unsigned)(i + d - RR) < (unsigned)NN;
    cv[d] = (unsigned)(j + d - RR) < (unsigned)NN;
  }

  float acc[WW];
  #pragma unroll
  for (int w = 0; w < WW; ++w) acc[w] = 0.0f;

  const int mbase = b * NPIX;

  for (int k = k0; k < k1; ++k) {
    const float* plane = feats + (size_t)(b * KK + k) * NPIX;
    // cooperative halo load: 484 entries covered by threads t and t+256
    #pragma unroll
    for (int s = 0; s < 2; ++s) {
      int idx = t + s * 256;
      if (idx < HALO * HALO) {
        int gi = i0 - RR + idx / HALO;
        int gj = j0 - RR + idx % HALO;
        float v = 0.0f;
        if ((unsigned)gi < (unsigned)NN && (unsigned)gj < (unsigned)NN) {
          float mv = (float)mask[mbase + gi * NN + gj];
          v = plane[gi * NN + gj] * mv;     // masked feature xm
        }
        tile[idx] = make_float2(fexp(v), fexp(-v));
      }
    }
    __syncthreads();

    const float2 ec = tile[(ty + RR) * HALO + (tx + RR)];
    float Z = 0.0f;
    #pragma unroll
    for (int di = 0; di < 7; ++di) {
      #pragma unroll
      for (int dj = 0; dj < 7; ++dj) {
        float2 nb = tile[(ty + di) * HALO + (tx + dj)];
        float e = fmaxf(ec.x * nb.y, ec.y * nb.x);   // exp(|x - nbr|)
        Z += (rv[di] && cv[dj]) ? e : 0.0f;
      }
    }
    const float coef = convp[k] / Z;
    #pragma unroll
    for (int di = 0; di < 7; ++di) {
      #pragma unroll
      for (int dj = 0; dj < 7; ++dj) {
        float2 nb = tile[(ty + di) * HALO + (tx + dj)];
        float e = fmaxf(ec.x * nb.y, ec.y * nb.x);
        acc[di * 7 + dj] += (rv[di] && cv[dj]) ? coef * e : 0.0f;
      }
    }
    __syncthreads();
  }

  const float m = (float)mask[mbase + i * NN + j];
  float* out = A + (size_t)((b * NN + i) * NN + j) * WW;
  #pragma unroll
  for (int w = 0; w < WW; ++w)
    unsafeAtomicAdd(&out[w], acc[w] * m);   // global_atomic_add_f32
}

// ---------------------------------------------------------------------------
// Pass 2: y(c, q) = sum_p A(p, w(p->q)) * x2(c, p)   via WMMA f32 16x16x4.
// One wave handles 16 output pixels (row i, cols j0..j0+15).
// For each source row r = i+di-3:  Y[16x21] += M[16x22] * X[22x21]
//   M[t,s] = A(b, r, j0-3+s, (6-di)*7 + (6-(s-t)))  for 0 <= s-t <= 6
//   X[s,c] = x2(b, c, r, j0-3+s)
// ---------------------------------------------------------------------------
__global__ __launch_bounds__(256)
void rwn_pass2(const float* __restrict__ A,    // [B][N][N][W]
               const float* __restrict__ x2,   // [B][C][N*N]
               float*       __restrict__ out)  // [B][N*N][C]
{
  const int lane   = threadIdx.x & 31;
  const int wave   = threadIdx.x >> 5;
  const int tileId = blockIdx.x * 8 + wave;     // 2048 tiles total
  const int b   = tileId >> 10;
  const int rem = tileId & 1023;
  const int i   = rem >> 3;
  const int j0  = (rem & 7) << 4;

  const int half = lane >> 4;     // 0: K=0/1 slice, 1: K=2/3 slice
  const int lm   = lane & 15;     // A-op: row t.  B-op: channel.  D: channel.

  v8f acc0 = {};   // channels 0..15
  v8f acc1 = {};   // channels 16..20 (lanes lm<5), rest padding

  for (int di = 0; di < 7; ++di) {
    const int r = i + di - RR;
    if ((unsigned)r >= (unsigned)NN) continue;          // wave-uniform skip
    const float* Arow   = A + (size_t)((b * NN + r) * NN) * WW;
    const int    x2base = b * CC * NPIX + r * NN;
    const int    wrow   = (6 - di) * 7;
    const int    c1     = (16 + lm < CC) ? (16 + lm) : (CC - 1);

    #pragma unroll
    for (int kk = 0; kk < 6; ++kk) {                    // K = 24 (22 used)
      v2f av, b0v, b1v;
      #pragma unroll
      for (int v = 0; v < 2; ++v) {
        const int s   = 4 * kk + 2 * half + v;          // source-halo column
        const int col = j0 - RR + s;
        const int d   = s - lm;                         // s - t
        // branchless: clamp addresses into range, select 0 after the load
        const bool aok = (s < HALO) & ((unsigned)d <= 6u) &
                         ((unsigned)col < (unsigned)NN);
        const bool bok = (s < HALO) & ((unsigned)col < (unsigned)NN);
        const int colc = (col < 0) ? 0 : ((col > NN - 1) ? NN - 1 : col);
        const int dc   = (d < 0) ? 0 : ((d > 6) ? 6 : d);
        const float aval = Arow[(size_t)colc * WW + wrow + (6 - dc)];
        const float bb0  = x2[x2base + lm * NPIX + colc];
        const float bb1  = x2[x2base + c1 * NPIX + colc];
        av[v]  = aok ? aval : 0.0f;
        b0v[v] = bok ? bb0 : 0.0f;
        b1v[v] = (bok && (lm < CC - 16)) ? bb1 : 0.0f;
      }
      acc0 = __builtin_amdgcn_wmma_f32_16x16x4_f32(
          false, av, false, b0v, (short)0, acc0, false, false);
      acc1 = __builtin_amdgcn_wmma_f32_16x16x4_f32(
          false, av, false, b1v, (short)0, acc1, false, false);
    }
  }

  // D layout: lane l, vgpr v -> (M = v + 8*half, N = lm)
  #pragma unroll
  for (int v = 0; v < 8; ++v) {
    const int tRow = v + 8 * half;                       // output pixel in tile
    const size_t q = (size_t)b * NPIX + (size_t)i * NN + (j0 + tRow);
    out[q * CC + lm] = acc0[v];
    if (lm < CC - 16) out[q * CC + 16 + lm] = acc1[v];
  }
}

// ---------------------------------------------------------------------------
extern "C" void kernel_launch(void* const* d_in, const int* in_sizes, int n_in,
                              void* d_out, int out_size, void* d_ws, size_t ws_size,
                              hipStream_t stream) {
  (void)in_sizes; (void)n_in; (void)out_size; (void)ws_size;
  const float* feats = (const float*)d_in[0];   // [2][131][128][128] f32
  const float* x2    = (const float*)d_in[1];   // [2][21][16384]     f32
  const int*   mask  = (const int*)d_in[2];     // [2][128][128]      i32
  const float* convp = (const float*)d_in[3];   // [131]              f32
  float* out = (float*)d_out;                   // [2][16384][21]     f32
  float* Aws = (float*)d_ws;                    // needs 2*16384*49*4 = 6.42 MB

  // A has 2*16384*49 = 1,605,632 floats = 6272 * 256
  rwn_zero<<<6272, 256, 0, stream>>>(Aws);

  dim3 g1(NN / TILE, NN / TILE, 2 * NGRP);      // 1024 blocks, 8K waves
  rwn_pass1<<<g1, 256, 0, stream>>>(feats, mask, convp, Aws);

  rwn_pass2<<<256, 256, 0, stream>>>(Aws, x2, out);
}